// InceptionLayer_28999619182973
// MI455X (gfx1250) — compile-verified
//
#include <hip/hip_runtime.h>
#include <math.h>

typedef __attribute__((ext_vector_type(2))) float v2f;
typedef __attribute__((ext_vector_type(8))) float v8f;

#define TPB 256

static __device__ __forceinline__ float leaky(float v, float s){ return v > 0.0f ? v : s * v; }

// ---------------- generic fills ----------------
__global__ void fill_f32_kernel(float* p, float v, long long n){
  long long i = blockIdx.x * (long long)blockDim.x + threadIdx.x;
  if (i < n) p[i] = v;
}
__global__ void fill_u32_kernel(unsigned int* p, unsigned int v, long long n){
  long long i = blockIdx.x * (long long)blockDim.x + threadIdx.x;
  if (i < n) p[i] = v;
}

// ---------------- degree + edge weights ----------------
__global__ void deg_kernel(const int* __restrict__ src, float* __restrict__ deg, int E){
  int e = blockIdx.x * blockDim.x + threadIdx.x;
  if (e < E) atomicAdd(&deg[src[e]], 1.0f);
}
__global__ void edgew_kernel(const int* __restrict__ src, const int* __restrict__ dst,
                             const float* __restrict__ deg, float* __restrict__ w, int E){
  int e = blockIdx.x * blockDim.x + threadIdx.x;
  if (e >= E) return;
  float di = deg[src[e]] * deg[dst[e]];
  w[e] = (di > 0.0f) ? -rsqrtf(fmaxf(di, 1e-12f)) : 0.0f;
}

// ---------------- WMMA GEMM: Out[M x NC] (op)= A[M x 32] * B[32 x NC] ----------------
// B (32 x NC, NC<=64) is staged once per workgroup into LDS.
// mode 0: overwrite, mode 1: accumulate into Out, mode 2: init from bias[col]
__global__ void __launch_bounds__(256) gemm_k32_kernel(
    const float* __restrict__ A, const float* __restrict__ B,
    float* __restrict__ Out, const float* __restrict__ bias,
    int M, int NC, int mode){
  __shared__ float Bs[32 * 64];
  // cooperative stage of B into LDS (all threads, before any exit)
  for (int i = threadIdx.x; i < 32 * NC; i += blockDim.x) Bs[i] = B[i];
  __syncthreads();

  int wave = blockIdx.x * (blockDim.x >> 5) + (threadIdx.x >> 5);
  int row0 = wave << 4;
  if (row0 >= M) return;                 // wave-uniform exit, EXEC stays all-ones
  int lane   = threadIdx.x & 31;
  int laneLo = lane & 15;
  int laneHi = lane >> 4;
#if __has_builtin(__builtin_amdgcn_wmma_f32_16x16x4_f32)
  // A fragment: 16x4 f32 per k-step; lanes 0-15 hold K={0,1}, lanes 16-31 K={2,3}
  const v2f* arow = (const v2f*)(A + (size_t)(row0 + laneLo) * 32 + 2 * laneHi);
  v2f a[8];
  #pragma unroll
  for (int kk = 0; kk < 8; ++kk) a[kk] = arow[kk * 2];   // 8-byte vector loads
  int ntiles = NC >> 4;
  for (int nt = 0; nt < ntiles; ++nt){
    int col = (nt << 4) + laneLo;
    const float* brow = Bs + (size_t)(2 * laneHi) * NC + col;
    v8f c;
    if (mode == 1){
      #pragma unroll
      for (int r = 0; r < 8; ++r) c[r] = Out[(size_t)(row0 + r + 8 * laneHi) * NC + col];
    } else if (mode == 2){
      float bv = bias[col];
      #pragma unroll
      for (int r = 0; r < 8; ++r) c[r] = bv;
    } else {
      #pragma unroll
      for (int r = 0; r < 8; ++r) c[r] = 0.0f;
    }
    #pragma unroll
    for (int kk = 0; kk < 8; ++kk){
      v2f bf;
      bf.x = brow[(kk * 4) * NC];
      bf.y = brow[(kk * 4 + 1) * NC];
      c = __builtin_amdgcn_wmma_f32_16x16x4_f32(false, a[kk], false, bf,
                                                (short)0, c, false, false);
    }
    #pragma unroll
    for (int r = 0; r < 8; ++r) Out[(size_t)(row0 + r + 8 * laneHi) * NC + col] = c[r];
  }
#else
  // scalar fallback (same tile ownership)
  int ntiles = NC >> 4;
  for (int nt = 0; nt < ntiles; ++nt){
    int col = (nt << 4) + laneLo;
    #pragma unroll
    for (int r = 0; r < 8; ++r){
      int row = row0 + r + 8 * laneHi;
      float acc = (mode == 1) ? Out[(size_t)row * NC + col]
                : (mode == 2) ? bias[col] : 0.0f;
      for (int k = 0; k < 32; ++k)
        acc += A[(size_t)row * 32 + k] * Bs[(size_t)k * NC + col];
      Out[(size_t)row * NC + col] = acc;
    }
  }
#endif
}

// ---------------- SpMV: out[dst] += w * x[src] (per edge, 32 channels) ----------------
__global__ void spmv_kernel(const float* __restrict__ x, const int* __restrict__ src,
                            const int* __restrict__ dst, const float* __restrict__ w,
                            float* __restrict__ out, int E){
  long long gid = blockIdx.x * (long long)blockDim.x + threadIdx.x;
  int e = (int)(gid >> 5);
  if (e >= E) return;
  int c = (int)(gid & 31);
  float v = w[e] * x[(size_t)src[e] * 32 + c];
  atomicAdd(&out[(size_t)dst[e] * 32 + c], v);
}

// Chebyshev recurrence: p = 2*p - q
__global__ void axpby_kernel(float* __restrict__ p, const float* __restrict__ q, long long n){
  long long i = blockIdx.x * (long long)blockDim.x + threadIdx.x;
  if (i < n) p[i] = 2.0f * p[i] - q[i];
}

// ---------------- SuperGAT edge passes ----------------
__global__ void gat_pass1(const float* __restrict__ h, const int* __restrict__ src,
                          const int* __restrict__ dst, const float* __restrict__ al,
                          const float* __restrict__ ar, float* __restrict__ abuf,
                          unsigned int* __restrict__ amax, int E, int N){
  long long gid = blockIdx.x * (long long)blockDim.x + threadIdx.x;
  long long idx = gid >> 1;
  if (idx >= (long long)E + N) return;
  int hd = (int)(gid & 1);
  int s = (idx < E) ? src[idx] : (int)(idx - E);
  int d = (idx < E) ? dst[idx] : (int)(idx - E);
  const float4* hj  = (const float4*)(h + ((size_t)s * 2 + hd) * 32);
  const float4* hi  = (const float4*)(h + ((size_t)d * 2 + hd) * 32);
  const float4* alv = (const float4*)(al + hd * 32);
  const float4* arv = (const float4*)(ar + hd * 32);
  float lg = 0.0f, aa = 0.0f;
  #pragma unroll
  for (int q = 0; q < 8; ++q){
    float4 j = hj[q], i = hi[q], l = alv[q], r = arv[q];
    lg += j.x * i.x + j.y * i.y + j.z * i.z + j.w * i.w;
    aa += j.x * l.x + j.y * l.y + j.z * l.z + j.w * l.w
        + i.x * r.x + i.y * r.y + i.z * r.z + i.w * r.w;
  }
  float sig = 1.0f / (1.0f + expf(-lg));
  float av  = leaky(aa * sig, 0.2f);
  abuf[idx * 2 + hd] = av;
  unsigned int u = __float_as_uint(av);
  u = (u & 0x80000000u) ? ~u : (u | 0x80000000u);     // order-preserving encode
  atomicMax(&amax[(size_t)d * 2 + hd], u);
}

__global__ void gat_pass2(const int* __restrict__ dst, const unsigned int* __restrict__ amax,
                          float* __restrict__ abuf, float* __restrict__ den, int E, int N){
  long long gid = blockIdx.x * (long long)blockDim.x + threadIdx.x;
  long long idx = gid >> 1;
  if (idx >= (long long)E + N) return;
  int hd = (int)(gid & 1);
  int d = (idx < E) ? dst[idx] : (int)(idx - E);
  unsigned int u = amax[(size_t)d * 2 + hd];
  float mx = (u & 0x80000000u) ? __uint_as_float(u & 0x7FFFFFFFu) : __uint_as_float(~u);
  float ex = expf(abuf[idx * 2 + hd] - mx);
  abuf[idx * 2 + hd] = ex;
  atomicAdd(&den[(size_t)d * 2 + hd], ex);
}

__global__ void gat_pass3(const float* __restrict__ h, const int* __restrict__ src,
                          const int* __restrict__ dst, const float* __restrict__ abuf,
                          const float* __restrict__ den, float* __restrict__ sgat,
                          int E, int N){
  long long gid = blockIdx.x * (long long)blockDim.x + threadIdx.x;
  long long idx = gid >> 6;
  if (idx >= (long long)E + N) return;
  int hc = (int)(gid & 63);
  int hd = hc >> 5;
  int s = (idx < E) ? src[idx] : (int)(idx - E);
  int d = (idx < E) ? dst[idx] : (int)(idx - E);
  float alpha = abuf[idx * 2 + hd] / (den[(size_t)d * 2 + hd] + 1e-16f);
  atomicAdd(&sgat[(size_t)d * 64 + hc], alpha * h[(size_t)s * 64 + hc]);
}

// ---------------- head-mean + residual leaky + BN stats ----------------
__global__ void __launch_bounds__(256) y_stats_kernel(
    const float* __restrict__ sgat, const float* __restrict__ gb,
    float* __restrict__ ybuf, float* __restrict__ stats, int N){
  __shared__ float sh[512];
  int c  = threadIdx.x & 31;
  int rl = threadIdx.x >> 5;              // 8 row-lanes per block
  float s = 0.0f, ss = 0.0f;
  for (int n = blockIdx.x * 8 + rl; n < N; n += gridDim.x * 8){
    float a0 = sgat[(size_t)n * 64 + c];
    float a1 = sgat[(size_t)n * 64 + 32 + c];
    float sm = 0.5f * (a0 + a1) + gb[c];
    float y  = leaky(sm, 0.01f) + sm;
    ybuf[(size_t)n * 32 + c] = y;
    s += y; ss += y * y;
  }
  sh[threadIdx.x] = s; sh[256 + threadIdx.x] = ss;
  __syncthreads();
  if (threadIdx.x < 32){
    float ts = 0.0f, tss = 0.0f;
    #pragma unroll
    for (int r = 0; r < 8; ++r){ ts += sh[r * 32 + c]; tss += sh[256 + r * 32 + c]; }
    atomicAdd(&stats[c], ts);
    atomicAdd(&stats[32 + c], tss);
  }
}

__global__ void bn_finalize_kernel(const float* __restrict__ stats,
                                   const float* __restrict__ gm, const float* __restrict__ bt,
                                   float* __restrict__ ssh, int N){
  int c = threadIdx.x;
  if (c < 32){
    float mu  = stats[c] / (float)N;
    float var = stats[32 + c] / (float)N - mu * mu;
    float sc  = gm[c] * rsqrtf(var + 1e-5f);
    ssh[c]      = sc;
    ssh[32 + c] = bt[c] - sc * mu;
  }
}

__global__ void bn_apply_kernel(const float* __restrict__ ybuf, const float* __restrict__ ssh,
                                float* __restrict__ acc, long long n){
  long long i = blockIdx.x * (long long)blockDim.x + threadIdx.x;
  if (i >= n) return;
  int c = (int)(i & 31);
  acc[i] += ssh[c] * ybuf[i] + ssh[32 + c];
}

__global__ void final_leaky_kernel(const float* __restrict__ acc, float* __restrict__ out, long long n){
  long long i = blockIdx.x * (long long)blockDim.x + threadIdx.x;
  if (i < n) out[i] = leaky(acc[i], 0.01f);
}

// ---------------- host orchestration ----------------
static inline unsigned int grd(long long n){ return (unsigned int)((n + TPB - 1) / TPB); }

extern "C" void kernel_launch(void* const* d_in, const int* in_sizes, int n_in,
                              void* d_out, int out_size, void* d_ws, size_t ws_size,
                              hipStream_t stream) {
  const int N = in_sizes[0] / 32;
  const int E = in_sizes[1] / 2;
  const float* x  = (const float*)d_in[0];
  const int*   ei = (const int*)d_in[1];
  const int*   src = ei;
  const int*   dst = ei + E;

  // workspace carve-up (floats)
  float* ws = (float*)d_ws;
  size_t o = 0;
  float* deg  = ws + o; o += (size_t)N;
  float* w    = ws + o; o += (size_t)E;
  float* b1   = ws + o; o += (size_t)N * 32;
  float* b2   = ws + o; o += (size_t)N * 32;
  float* b3   = ws + o; o += (size_t)N * 32;
  float* cheb = ws + o; o += (size_t)N * 32;
  float* hbuf = ws + o; o += (size_t)N * 64;
  float* abuf = ws + o; o += ((size_t)E + N) * 2;
  unsigned int* amax = (unsigned int*)(ws + o); o += (size_t)N * 2;
  float* den  = ws + o; o += (size_t)N * 2;
  float* sgat = ws + o; o += (size_t)N * 64;
  float* ybuf = ws + o; o += (size_t)N * 32;
  float* acc  = ws + o; o += (size_t)N * 32;
  float* stats= ws + o; o += 64;
  float* ssh  = ws + o; o += 64;

  const long long NC32 = (long long)N * 32;
  const long long Etot2 = ((long long)E + N) * 2;
  const long long Etot64 = ((long long)E + N) * 64;
  const unsigned int gemmGrid = (unsigned int)(((N + 15) / 16 + 7) / 8);

  // degree + normalized edge weights (shared across all blocks)
  fill_f32_kernel<<<grd(N), TPB, 0, stream>>>(deg, 0.0f, N);
  deg_kernel<<<grd(E), TPB, 0, stream>>>(src, deg, E);
  edgew_kernel<<<grd(E), TPB, 0, stream>>>(src, dst, deg, w, E);

  fill_f32_kernel<<<grd(NC32), TPB, 0, stream>>>(acc, 0.0f, NC32);

  float* pool[3] = { b1, b2, b3 };

  for (int b = 0; b < 4; ++b){
    const float* Wb = (const float*)d_in[3 + 8 * b + 0];
    const float* cb = (const float*)d_in[3 + 8 * b + 1];
    const float* G  = (const float*)d_in[3 + 8 * b + 2];
    const float* al = (const float*)d_in[3 + 8 * b + 3];
    const float* ar = (const float*)d_in[3 + 8 * b + 4];
    const float* gb = (const float*)d_in[3 + 8 * b + 5];
    const float* gm = (const float*)d_in[3 + 8 * b + 6];
    const float* bt = (const float*)d_in[3 + 8 * b + 7];
    const int K = in_sizes[3 + 8 * b] / (32 * 32);

    // ---- ChebConv ----
    // cheb = x @ W0 + cb
    gemm_k32_kernel<<<gemmGrid, TPB, 0, stream>>>(x, Wb, cheb, cb, N, 32, 2);
    // T1 = spmv(x)
    fill_f32_kernel<<<grd(NC32), TPB, 0, stream>>>(b1, 0.0f, NC32);
    spmv_kernel<<<grd((long long)E * 32), TPB, 0, stream>>>(x, src, dst, w, b1, E);
    gemm_k32_kernel<<<gemmGrid, TPB, 0, stream>>>(b1, Wb + 1024, cheb, nullptr, N, 32, 1);

    const float* tx0 = x;
    float* tx1 = b1;
    for (int k = 2; k < K; ++k){
      float* tn = pool[(k - 1) % 3];
      fill_f32_kernel<<<grd(NC32), TPB, 0, stream>>>(tn, 0.0f, NC32);
      spmv_kernel<<<grd((long long)E * 32), TPB, 0, stream>>>(tx1, src, dst, w, tn, E);
      axpby_kernel<<<grd(NC32), TPB, 0, stream>>>(tn, tx0, NC32);   // tn = 2*spmv - tx0
      gemm_k32_kernel<<<gemmGrid, TPB, 0, stream>>>(tn, Wb + (size_t)k * 1024, cheb, nullptr, N, 32, 1);
      tx0 = tx1; tx1 = tn;
    }

    // ---- SuperGAT ----
    // h = cheb @ G  -> [N, H*C]
    gemm_k32_kernel<<<gemmGrid, TPB, 0, stream>>>(cheb, G, hbuf, nullptr, N, 64, 0);
    fill_u32_kernel<<<grd((long long)N * 2), TPB, 0, stream>>>(amax, 0u, (long long)N * 2);
    fill_f32_kernel<<<grd((long long)N * 2), TPB, 0, stream>>>(den, 0.0f, (long long)N * 2);
    fill_f32_kernel<<<grd((long long)N * 64), TPB, 0, stream>>>(sgat, 0.0f, (long long)N * 64);

    gat_pass1<<<grd(Etot2), TPB, 0, stream>>>(hbuf, src, dst, al, ar, abuf, amax, E, N);
    gat_pass2<<<grd(Etot2), TPB, 0, stream>>>(dst, amax, abuf, den, E, N);
    gat_pass3<<<grd(Etot64), TPB, 0, stream>>>(hbuf, src, dst, abuf, den, sgat, E, N);

    // ---- mean heads + bias, residual leaky, BN ----
    fill_f32_kernel<<<1, 64, 0, stream>>>(stats, 0.0f, 64);
    y_stats_kernel<<<256, TPB, 0, stream>>>(sgat, gb, ybuf, stats, N);
    bn_finalize_kernel<<<1, 32, 0, stream>>>(stats, gm, bt, ssh, N);
    bn_apply_kernel<<<grd(NC32), TPB, 0, stream>>>(ybuf, ssh, acc, NC32);
  }

  final_leaky_kernel<<<grd(NC32), TPB, 0, stream>>>(acc, (float*)d_out, NC32);
}